// TripletLoss_44513041056155
// MI455X (gfx1250) — compile-verified
//
#include <hip/hip_runtime.h>
#include <math.h>

typedef __attribute__((ext_vector_type(2))) float v2f;
typedef __attribute__((ext_vector_type(8))) float v8f;

#define DIMD 128

// ---------------- Kernel 1: per-row squared norms (one wave32 per row) ----
__global__ __launch_bounds__(256)
void sqnorm_kernel(const float* __restrict__ x, float* __restrict__ sq, int N) {
    int gid  = blockIdx.x * blockDim.x + threadIdx.x;
    int row  = gid >> 5;
    int lane = threadIdx.x & 31;
    if (row >= N) return;
    const float4* rp = (const float4*)(x + (size_t)row * DIMD);
    float4 v = rp[lane];                       // 32 lanes x float4 = 128 floats
    float s = v.x * v.x + v.y * v.y + v.z * v.z + v.w * v.w;
#pragma unroll
    for (int off = 16; off > 0; off >>= 1) s += __shfl_xor(s, off, 32);
    if (lane == 0) sq[row] = s;
}

// ---------------- Kernel 2: WMMA triplet dot products + loss partials -----
// One wave handles 16 triplets. A rows = x[i_t], B cols = x[j_t] (then x[k_t]).
// D = A*B^T accumulated over K=128 via 32 chained V_WMMA_F32_16X16X4_F32;
// the 16 needed dot products are the diagonal of the 16x16 result.
__global__ __launch_bounds__(256)
void triplet_kernel(const float* __restrict__ x, const int* __restrict__ trip,
                    const float* __restrict__ sq, float* __restrict__ partial,
                    int T) {
    __shared__ float sdij[8][16];
    __shared__ float sdik[8][16];
    __shared__ float wsum[8];

    const int lane = threadIdx.x & 31;
    const int w    = threadIdx.x >> 5;          // wave in block (8 waves)
    const long long tile  = (long long)blockIdx.x * 8 + w;
    const long long tbase = tile * 16;

    const int  r     = lane & 15;               // triplet slot / matrix row+col
    const long long tt = tbase + r;
    const bool valid = (tt < (long long)T);
    const int  tcl   = valid ? (int)tt : (T - 1);   // clamp: keep EXEC all-1s

    const int i = trip[3 * tcl + 0];
    const int j = trip[3 * tcl + 1];
    const int k = trip[3 * tcl + 2];

    const float* xi = x + (size_t)i * DIMD;
    const float* xj = x + (size_t)j * DIMD;
    const float* xk = x + (size_t)k * DIMD;

    // ISA 16x4 f32 A-layout: lanes 0-15 hold K={0,1}, lanes 16-31 hold K={2,3}
    const int koff = (lane >> 4) << 1;          // 0 or 2

    v8f cij = {};
    v8f cik = {};
#pragma unroll
    for (int kk = 0; kk < DIMD; kk += 4) {
        v2f a  = *(const v2f*)(xi + kk + koff);
        v2f bj = *(const v2f*)(xj + kk + koff);
        v2f bk = *(const v2f*)(xk + kk + koff);
        // (neg_a, A, neg_b, B, c_mod, C, reuse_a, reuse_b)
        cij = __builtin_amdgcn_wmma_f32_16x16x4_f32(false, a, false, bj,
                                                    (short)0, cij, false, false);
        cik = __builtin_amdgcn_wmma_f32_16x16x4_f32(false, a, false, bk,
                                                    (short)0, cik, false, false);
    }

    // Diagonal extraction: C/D layout -> lanes 0-7 hold the diagonal entry of row lane in result element lane,
    // lanes 24-31 hold the diagonal entry of row lane-16 in result element lane-24.
    int vsel = -1, tl = 0;
    if (lane < 8)        { vsel = lane;      tl = lane;      }
    else if (lane >= 24) { vsel = lane - 24; tl = lane - 16; }
    float dj = cij[0], dk = cik[0];
#pragma unroll
    for (int v = 1; v < 8; ++v) {
        if (vsel == v) { dj = cij[v]; dk = cik[v]; }
    }
    if (vsel >= 0) { sdij[w][tl] = dj; sdik[w][tl] = dk; }
    __syncthreads();

    float term = 0.0f;
    if (lane < 16) {                            // this lane's triplet is (i,j,k)
        float dotij = sdij[w][lane];
        float dotik = sdik[w][lane];
        float sqi = sq[i], sqj = sq[j], sqk = sq[k];
        float dij = fmaxf(sqi + sqj - 2.0f * dotij, 0.0f);
        float dik = fmaxf(sqi + sqk - 2.0f * dotik, 0.0f);
        // naive (reference-matching) softplus: overflows to +inf exactly as jnp
        term = valid ? log1pf(expf(dij - dik)) : 0.0f;
    }
#pragma unroll
    for (int off = 16; off > 0; off >>= 1) term += __shfl_xor(term, off, 32);
    if (lane == 0) wsum[w] = term;
    __syncthreads();

    if (threadIdx.x == 0) {
        float s = 0.0f;
#pragma unroll
        for (int q = 0; q < 8; ++q) s += wsum[q];
        partial[blockIdx.x] = s;
    }
}

// ---------------- Kernel 3: deterministic final reduction -----------------
__global__ __launch_bounds__(256)
void finalize_kernel(const float* __restrict__ partial, int nPart,
                     float* __restrict__ out, float invT) {
    __shared__ float red[256];
    float s = 0.0f;
    for (int idx = threadIdx.x; idx < nPart; idx += 256) s += partial[idx];
    red[threadIdx.x] = s;
    __syncthreads();
    for (int off = 128; off > 0; off >>= 1) {
        if ((int)threadIdx.x < off) red[threadIdx.x] += red[threadIdx.x + off];
        __syncthreads();
    }
    if (threadIdx.x == 0) out[0] = red[0] * invT;
}

// ---------------- Host-side launch ----------------------------------------
extern "C" void kernel_launch(void* const* d_in, const int* in_sizes, int n_in,
                              void* d_out, int out_size, void* d_ws, size_t ws_size,
                              hipStream_t stream) {
    const float* x    = (const float*)d_in[0];
    const int*   trip = (const int*)d_in[1];

    const int N = in_sizes[0] / DIMD;   // 8192 rows
    const int T = in_sizes[1] / 3;      // 200000 triplets

    float* sq      = (float*)d_ws;      // N floats
    float* partial = sq + N;            // nBlocks floats

    const int nTiles  = (T + 15) / 16;          // 16 triplets per wave
    const int nBlocks = (nTiles + 7) / 8;       // 8 waves per 256-thread block
    const int sqBlocks = (N + 7) / 8;           // 8 rows (waves) per block

    sqnorm_kernel<<<sqBlocks, 256, 0, stream>>>(x, sq, N);
    triplet_kernel<<<nBlocks, 256, 0, stream>>>(x, trip, sq, partial, T);
    finalize_kernel<<<1, 256, 0, stream>>>(partial, nBlocks, (float*)d_out,
                                           1.0f / (float)T);
}